// BatchDepthwiseCrossCorrelation_14542759264966
// MI455X (gfx1250) — compile-verified
//
#include <hip/hip_runtime.h>

// Shapes hardcoded per reference:
//   x:         [BS=8, NC=256, HI=64, WI=64]  f32
//   templates: [NT=8, BS=8, NC=256, 7, 7]    f32
//   out:       [NT=8, BS=8, NC=256, 64, 64]  f32, same-padding (pad=3)

typedef float v2f __attribute__((ext_vector_type(2)));
typedef float v8f __attribute__((ext_vector_type(8)));

#define HI     64
#define WI     64
#define HT     7
#define WT     7
#define PAD    3
#define PH     70            // padded image dim (64 + 2*3)
#define NT     8
#define NCH    2048          // BS*NC
#define KTOT   49            // 7*7
#define KSTEPS 13            // ceil(49/4); K padded to 52
#define TILES_PER_WAVE 32    // 256 tiles / 8 waves

__global__ __launch_bounds__(256)
void dwxcorr_wmma_f32(const float* __restrict__ x,
                      const float* __restrict__ tmpl,
                      float* __restrict__ out)
{
    __shared__ float img[PH * PH];      // 19600 B zero-padded channel image

    const int ch   = blockIdx.x;        // 0..2047 == b*256 + c
    const int tid  = threadIdx.x;       // 0..255
    const int lane = tid & 31;
    const int wave = tid >> 5;          // 0..7

    // ---- stage the padded channel image into LDS -------------------------
    const float* xc = x + (size_t)ch * (HI * WI);
    for (int i = tid; i < PH * PH; i += 256) {
        int yy = i / PH - PAD;
        int xx = i % PH - PAD;
        float v = 0.0f;
        if (yy >= 0 && yy < HI && xx >= 0 && xx < WI) v = xc[yy * WI + xx];
        img[i] = v;
    }

    // ---- lane roles per WMMA f32 16x16x4 layouts -------------------------
    // A (16x4): lane L holds M = L&15; VGPR0 -> K = k0 + (L>=16 ? 2:0), VGPR1 -> +1
    // B (4x16): lane L holds N = L&15; VGPR0 -> row K = k0 + (L>=16 ? 2:0), VGPR1 -> +1
    // C (16x16): VGPR r: lanes 0-15 -> (M=r, N=L), lanes 16-31 -> (M=r+8, N=L-16)
    const int m     = lane & 15;              // A: pixel-in-tile, B/C: N index
    const int khalf = (lane >> 4) * 2;        // 0 for low half, 2 for high half

    // ---- load B fragments (templates) once per channel: 13 x 2 VGPRs ----
    float b0[KSTEPS], b1[KSTEPS];
    {
        const float* wbase = tmpl + ((size_t)m * NCH + ch) * (HT * WT);
        #pragma unroll
        for (int s = 0; s < KSTEPS; ++s) {
            int k = s * 4 + khalf;
            float v0 = 0.0f, v1 = 0.0f;
            if (m < NT) {
                if (k     < KTOT) v0 = wbase[k];
                if (k + 1 < KTOT) v1 = wbase[k + 1];
            }
            b0[s] = v0;                                    // zero-padded K/N
            b1[s] = v1;
        }
    }

    // ---- per-lane LDS offsets for A fragments (clamped; b==0 covers pad) -
    int offA0[KSTEPS], offA1[KSTEPS];
    #pragma unroll
    for (int s = 0; s < KSTEPS; ++s) {
        int k0  = s * 4 + khalf;
        int kc0 = k0     > 48 ? 48 : k0;
        int kc1 = k0 + 1 > 48 ? 48 : k0 + 1;
        offA0[s] = (kc0 / WT) * PH + (kc0 % WT) + m;
        offA1[s] = (kc1 / WT) * PH + (kc1 % WT) + m;
    }

    __syncthreads();

    // LDS base of a logical tile t (0..31) owned by this wave
    #define TILE_BASE(t) ((((((t) * 8) + wave) >> 2) * PH) + (((((t) * 8) + wave) & 3) * 16))

    // Batched A-fragment fetch: 26 ds_loads issued together (one wait point)
    #define LOAD_TILE(BASE, A0, A1)                          \
        { const int _b = (BASE);                             \
          _Pragma("unroll")                                  \
          for (int s = 0; s < KSTEPS; ++s) {                 \
              A0[s] = img[_b + offA0[s]];                    \
              A1[s] = img[_b + offA1[s]];                    \
          } }

    // 13 back-to-back WMMAs (accumulation chain), then contiguous b128 stores
    #define COMPUTE_STORE(T, A0, A1)                                           \
        { v8f c = {};                                                          \
          _Pragma("unroll")                                                    \
          for (int s = 0; s < KSTEPS; ++s) {                                   \
              v2f a, b;                                                        \
              a[0] = A0[s]; a[1] = A1[s];                                      \
              b[0] = b0[s]; b[1] = b1[s];                                      \
              c = __builtin_amdgcn_wmma_f32_16x16x4_f32(                       \
                      false, a, false, b, (short)0, c, false, false);          \
          }                                                                    \
          if (m < NT) {                                                        \
              const int _tile = (T) * 8 + wave;                                \
              const int _ty   = _tile >> 2;                                    \
              const int _tx   = (_tile & 3) * 16;                              \
              float* _op = out + ((size_t)m * NCH + ch) * (HI * WI)            \
                               + _ty * WI + _tx + ((lane < 16) ? 0 : 8);       \
              *(v8f*)_op = c;                                                  \
          } }

    // ---- software-pipelined tile loop (double-buffered A fragments) ------
    float pa0[KSTEPS], pa1[KSTEPS], qa0[KSTEPS], qa1[KSTEPS];
    LOAD_TILE(TILE_BASE(0), pa0, pa1);
    for (int i = 0; i < TILES_PER_WAVE; i += 2) {
        LOAD_TILE(TILE_BASE((i + 1) & 31), qa0, qa1);   // prefetch next
        COMPUTE_STORE(i, pa0, pa1);                     // consume current
        LOAD_TILE(TILE_BASE((i + 2) & 31), pa0, pa1);   // prefetch next+1 (wraps; harmless)
        COMPUTE_STORE(i + 1, qa0, qa1);
    }

    #undef TILE_BASE
    #undef LOAD_TILE
    #undef COMPUTE_STORE
}

extern "C" void kernel_launch(void* const* d_in, const int* in_sizes, int n_in,
                              void* d_out, int out_size, void* d_ws, size_t ws_size,
                              hipStream_t stream) {
    const float* x    = (const float*)d_in[0];
    const float* tmpl = (const float*)d_in[1];
    float*       out  = (float*)d_out;
    dwxcorr_wmma_f32<<<dim3(NCH), dim3(256), 0, stream>>>(x, tmpl, out);
}